// VQEmbedding_56281251447176
// MI455X (gfx1250) — compile-verified
//
#include <hip/hip_runtime.h>
#include <hip/hip_bf16.h>
#include <cstddef>

// ---------------------------------------------------------------------------
// VQ-VAE vector quantization for MI455X (gfx1250, wave32).
//   h:        [32768, 256] fp32      codebook: [1024, 256] fp32
//   out:      [32768*256] quantized  + out[NTOT]=commitment, out[NTOT+1]=cbloss
// Distance GEMM (17.2 GFLOP fp32) runs on V_WMMA_F32_16X16X4_F32 to keep the
// argmin numerically exact in fp32. A-operand is staged as (-2*h) in LDS so
// the accumulator is directly -2*h.W; adding ||W||^2 gives the distance.
// ---------------------------------------------------------------------------

typedef float v2f __attribute__((ext_vector_type(2)));
typedef float v8f __attribute__((ext_vector_type(8)));

#define NROWS   32768      // 32*1024 input vectors
#define DIM     256
#define NCODE   1024
#define NTOT    8388608    // 32768*256
#define ASTRIDE 260        // LDS row stride: 4*260 B -> 4-bank shift/row, conflict-free

// ---- Kernel 0: codebook row squared norms --------------------------------
__global__ __launch_bounds__(256) void vq_wnorm_kernel(
    const float* __restrict__ codebook, float* __restrict__ wnorm)
{
    int r = blockIdx.x * 256 + threadIdx.x;
    if (r < NCODE) {
        const float4* p = (const float4*)(codebook + (size_t)r * DIM);
        float s = 0.f;
        #pragma unroll 8
        for (int i = 0; i < DIM / 4; ++i) {
            float4 v = p[i];
            s += v.x * v.x + v.y * v.y + v.z * v.z + v.w * v.w;
        }
        wnorm[r] = s;
    }
}

// ---- Kernel 1: fused distance-GEMM + argmin (fp32 WMMA) ------------------
// Workgroup = 256 threads (8 waves). Each block owns 16 rows of h and scans
// all 1024 codebook entries. Wave w handles n-tiles [w*8, w*8+8).
__global__ __launch_bounds__(256) void vq_argmin_kernel(
    const float* __restrict__ h, const float* __restrict__ codebook,
    const float* __restrict__ wnorm, int* __restrict__ indices)
{
    __shared__ float As[16 * ASTRIDE];        // -2*h strip, 16 x 256
    __shared__ float sVal[8][32][8];
    __shared__ int   sIdx[8][32][8];

    const int tid     = threadIdx.x;
    const int rowBase = blockIdx.x * 16;

    // Stage A = -2*h (coalesced: each thread writes column tid of 16 rows).
    #pragma unroll
    for (int i = 0; i < 16; ++i)
        As[i * ASTRIDE + tid] = -2.0f * h[(size_t)(rowBase + i) * DIM + tid];
    __syncthreads();

    const int w    = tid >> 5;
    const int lane = tid & 31;
    const int lh   = lane & 15;              // A: row M / B: col N within tile
    const int kOff = (lane < 16) ? 0 : 2;    // K sub-pair per ISA 16x4 layout

    float minv[8];
    int   mini[8];
    #pragma unroll
    for (int j = 0; j < 8; ++j) { minv[j] = 3.4e38f; mini[j] = 0; }

    for (int t8 = 0; t8 < 8; ++t8) {
        const int nbase = (w * 8 + t8) * 16;
        const int n     = nbase + lh;
        const float* bp = codebook + (size_t)n * DIM + kOff;
        const float* ap = &As[lh * ASTRIDE + kOff];

        v8f acc = {};                        // 16x16 fp32 accumulator tile
        #pragma unroll 8
        for (int ck = 0; ck < DIM / 4; ++ck) {
            v2f a = *(const v2f*)(ap + ck * 4);   // LDS ds_load_b64
            v2f b = *(const v2f*)(bp + ck * 4);   // global_load_b64 (L2-hot)
            acc = __builtin_amdgcn_wmma_f32_16x16x4_f32(
                false, a, false, b, (short)0, acc, false, false);
        }

        const float wn = wnorm[n];
        // C layout: VGPR j holds row (lane<16 ? j : j+8), column n.
        #pragma unroll
        for (int j = 0; j < 8; ++j) {
            float d = wn + acc[j];           // ||W||^2 - 2 h.W  (drop ||h||^2)
            if (d < minv[j]) { minv[j] = d; mini[j] = n; }
        }
    }

    #pragma unroll
    for (int j = 0; j < 8; ++j) {
        sVal[w][lane][j] = minv[j];
        sIdx[w][lane][j] = mini[j];
    }
    __syncthreads();

    // Final per-row reduction: thread t (<16) owns row t of this strip.
    if (tid < 16) {
        const int j       = tid & 7;
        const int laneOfs = (tid < 8) ? 0 : 16;
        float best = 3.4e38f;
        int   bi   = 0;
        for (int w2 = 0; w2 < 8; ++w2)
            for (int l = 0; l < 16; ++l) {
                float v = sVal[w2][laneOfs + l][j];
                int   i = sIdx[w2][laneOfs + l][j];
                if (v < best || (v == best && i < bi)) { best = v; bi = i; }
            }
        indices[rowBase + tid] = bi;
    }
}

// ---- Kernel 2: gather codebook rows + per-block SSE partials -------------
// One wave per row (8 rows / block). Deterministic block reduction.
__global__ __launch_bounds__(256) void vq_gather_kernel(
    const float* __restrict__ h, const float* __restrict__ codebook,
    const int* __restrict__ indices, float* __restrict__ out,
    float* __restrict__ partials)
{
    __shared__ float sP[256];
    const int tid  = threadIdx.x;
    const int w    = tid >> 5;
    const int lane = tid & 31;
    const int row  = blockIdx.x * 8 + w;
    const int idx  = indices[row];

    const float4* hp = (const float4*)h;
    const float4* cp = (const float4*)codebook;
    float4*       op = (float4*)out;

    float s = 0.f;
    #pragma unroll
    for (int i = 0; i < 2; ++i) {
        const int c4 = lane + 32 * i;              // 64 float4 per row
        float4 q  = cp[(size_t)idx * 64 + c4];
        float4 hv = hp[(size_t)row * 64 + c4];
        op[(size_t)row * 64 + c4] = q;
        float dx = hv.x - q.x, dy = hv.y - q.y;
        float dz = hv.z - q.z, dw = hv.w - q.w;
        s += dx * dx + dy * dy + dz * dz + dw * dw;
    }

    sP[tid] = s;
    __syncthreads();
    for (int st = 128; st > 0; st >>= 1) {
        if (tid < st) sP[tid] += sP[tid + st];
        __syncthreads();
    }
    if (tid == 0) partials[blockIdx.x] = sP[0];
}

// ---- Kernel 3: finalize losses -------------------------------------------
__global__ __launch_bounds__(256) void vq_finalize_kernel(
    const float* __restrict__ partials, float* __restrict__ out)
{
    __shared__ float sP[256];
    const int tid = threadIdx.x;
    float s = 0.f;
    for (int i = tid; i < 4096; i += 256) s += partials[i];
    sP[tid] = s;
    __syncthreads();
    for (int st = 128; st > 0; st >>= 1) {
        if (tid < st) sP[tid] += sP[tid + st];
        __syncthreads();
    }
    if (tid == 0) {
        const float sse  = sP[0];
        const float invN = 1.0f / (float)NTOT;
        out[NTOT]     = 0.25f * sse * invN;   // commitment_loss
        out[NTOT + 1] = sse * invN;           // codebook_loss (same forward value)
    }
}

// ---------------------------------------------------------------------------
extern "C" void kernel_launch(void* const* d_in, const int* in_sizes, int n_in,
                              void* d_out, int out_size, void* d_ws, size_t ws_size,
                              hipStream_t stream)
{
    const float* h  = (const float*)d_in[0];   // [32,1024,256] fp32
    const float* cb = (const float*)d_in[1];   // [1024,256] fp32
    float*       out = (float*)d_out;          // [NTOT + 2] fp32

    // Workspace layout (floats): wnorm[1024] | partials[4096] | indices[32768]
    float* wsf      = (float*)d_ws;
    float* wnorm    = wsf;
    float* partials = wsf + 1024;
    int*   indices  = (int*)(wsf + 1024 + 4096);

    vq_wnorm_kernel  <<<(NCODE + 255) / 256, 256, 0, stream>>>(cb, wnorm);
    vq_argmin_kernel <<<NROWS / 16,          256, 0, stream>>>(h, cb, wnorm, indices);
    vq_gather_kernel <<<NROWS / 8,           256, 0, stream>>>(h, cb, indices, out, partials);
    vq_finalize_kernel<<<1,                  256, 0, stream>>>(partials, out);
}